// OpenMaxLayer_35837207118016
// MI455X (gfx1250) — compile-verified
//
#include <hip/hip_runtime.h>
#include <math.h>

// Problem constants (match reference)
#define B_ROWS 16384
#define C_CLS  1000
#define C_PAD  1008   // 63 tiles of 16
#define F_DIM  256
#define LDA    260    // padded LDS stride for features tile (bank-conflict-free)
#define SC_STR 1009   // padded LDS stride for score tile

typedef float v2f __attribute__((ext_vector_type(2)));
typedef float v8f __attribute__((ext_vector_type(8)));

// ---------------------------------------------------------------------------
// Kernel 0: build mT[F_DIM][C_PAD] = mean_vecs^T (zero padded cols) and a
// per-class float4 parameter table: {||m_c||^2, shape_c, loc_c, 1/scale_c}.
// grid = C_PAD blocks, block = 256 threads (one per k)
// ---------------------------------------------------------------------------
__global__ void openmax_prep(const float* __restrict__ mean_vecs,
                             const float* __restrict__ wb_shape,
                             const float* __restrict__ wb_loc,
                             const float* __restrict__ wb_scale,
                             float* __restrict__ mT,
                             float4* __restrict__ params) {
    const int c = blockIdx.x;    // 0..1007
    const int k = threadIdx.x;   // 0..255
    float v = 0.0f;
    if (c < C_CLS) v = mean_vecs[(size_t)c * F_DIM + k];
    mT[(size_t)k * C_PAD + c] = v;

    __shared__ float red[256];
    red[k] = v * v;
    __syncthreads();
    for (int s = 128; s > 0; s >>= 1) {
        if (k < s) red[k] += red[k + s];
        __syncthreads();
    }
    if (k == 0 && c < C_CLS) {
        params[c] = make_float4(red[0], wb_shape[c], wb_loc[c],
                                1.0f / wb_scale[c]);
    }
}

// ---------------------------------------------------------------------------
// Kernel 1: fused distance-GEMM (f32 WMMA) + Weibull reweight + row softmax.
// grid = B_ROWS/16 blocks, block = 128 threads (4 wave32s).
// Each block: 16 rows x all 1000 classes; scores live only in LDS.
// ---------------------------------------------------------------------------
__global__ __launch_bounds__(128)
void openmax_main(const float* __restrict__ logits,
                  const float* __restrict__ features,
                  const float* __restrict__ mT,
                  const float4* __restrict__ params,
                  float* __restrict__ out) {
    __shared__ float Af[16 * LDA];        // features tile, padded stride
    __shared__ float Sc[16 * SC_STR];     // score tile (incl. pad cols)
    __shared__ float f2[16];
    __shared__ float pred[8 * 16];        // partial reduce scratch
    __shared__ float rowred[16];          // row max, then 1/rowsum

    const int tid  = threadIdx.x;
    const int lane = tid & 31;
    const int wave = tid >> 5;            // 0..3
    const int half = (lane >> 4) & 1;     // 0: lanes 0-15, 1: lanes 16-31
    const int ln   = lane & 15;
    const int r0   = blockIdx.x * 16;     // first global row of this tile

    // --- stage features tile (16x256 f32) into LDS, coalesced float4 ---
    {
        const float4* src = (const float4*)(features + (size_t)r0 * F_DIM);
        #pragma unroll
        for (int j = 0; j < 8; ++j) {
            int v4   = tid + j * 128;     // 0..1023
            float4 d = src[v4];
            int flat = v4 * 4;
            int r    = flat >> 8;         // /256
            int col  = flat & 255;
            *(float4*)&Af[r * LDA + col] = d;
        }
    }
    __syncthreads();

    // --- f2[r] = ||features_row||^2 ---
    if (tid < 16) {
        float s = 0.0f;
        for (int k = 0; k < F_DIM; ++k) { float x = Af[tid * LDA + k]; s += x * x; }
        f2[tid] = s;
    }
    __syncthreads();

    // --- per-wave N-tile loop: 63 tiles of 16 classes ---
    for (int t = wave; t < 63; t += 4) {
        const int n0 = t * 16;
        v8f acc0 = {};                     // two accumulators break the
        v8f acc1 = {};                     // WMMA->WMMA RAW chain
        const float* ap = Af + ln * LDA + 2 * half;                   // A frag
        const float* bp = mT + n0 + ln + (size_t)(2 * half) * C_PAD;  // B frag

        #pragma unroll 4
        for (int k = 0; k < F_DIM; k += 8) {
            v2f a0, b0, a1, b1;
            a0.x = ap[k];
            a0.y = ap[k + 1];
            b0.x = bp[(size_t)k * C_PAD];
            b0.y = bp[(size_t)k * C_PAD + C_PAD];
            a1.x = ap[k + 4];
            a1.y = ap[k + 5];
            b1.x = bp[(size_t)(k + 4) * C_PAD];
            b1.y = bp[(size_t)(k + 4) * C_PAD + C_PAD];
            acc0 = __builtin_amdgcn_wmma_f32_16x16x4_f32(
                false, a0, false, b0, (short)0, acc0, false, false);
            acc1 = __builtin_amdgcn_wmma_f32_16x16x4_f32(
                false, a1, false, b1, (short)0, acc1, false, false);
        }

        // --- branchless epilogue: d2 -> dist -> weibull CDF -> score ---
        const int   n     = n0 + ln;
        const bool  valid = (n < C_CLS);
        const int   nc    = valid ? n : (C_CLS - 1);
        const float4 p    = params[nc];           // {m2, shape, loc, 1/scale}
        const float m2v = p.x, shp = p.y, loc = p.z, iscale = p.w;
        const float* lrow = logits + (size_t)r0 * C_CLS + nc;

        #pragma unroll
        for (int g = 0; g < 8; ++g) {
            const int ml   = g + 8 * half;        // local row 0..15
            float dot  = acc0[g] + acc1[g];
            float d2   = f2[ml] + m2v - 2.0f * dot;
            float dist = sqrtf(fmaxf(d2, 1e-12f));
            float xp   = (dist - loc) * iscale;
            // xp^shape via exp/log; selected away when xp <= 0
            float pw   = __expf(shp * __logf(xp));
            float w    = (xp > 0.0f) ? (1.0f - __expf(-pw)) : 0.0f;
            float w2   = w * w;
            float w4   = w2 * w2;
            float w8   = w4 * w4;
            float w10  = w8 * w2;                 // w^ALPHA, ALPHA=10
            float s    = lrow[(size_t)ml * C_CLS] * (1.0f - w10);
            s = valid ? s : -1e30f;               // padded class -> -inf-ish
            Sc[ml * SC_STR + n] = s;              // unconditional store
        }
    }
    __syncthreads();

    // --- workgroup softmax over C=1000 per row ---
    const int sub = tid >> 4;    // 0..7 column-strided worker
    const int row = tid & 15;

    float lmax = -1e30f;
    for (int c = sub; c < C_PAD; c += 8)
        lmax = fmaxf(lmax, Sc[row * SC_STR + c]);
    pred[sub * 16 + row] = lmax;
    __syncthreads();
    if (tid < 16) {
        float m = -1e30f;
        #pragma unroll
        for (int s = 0; s < 8; ++s) m = fmaxf(m, pred[s * 16 + tid]);
        rowred[tid] = m;
    }
    __syncthreads();

    const float rm = rowred[row];
    float lsum = 0.0f;
    for (int c = sub; c < C_CLS; c += 8) {
        float e = __expf(Sc[row * SC_STR + c] - rm);
        Sc[row * SC_STR + c] = e;
        lsum += e;
    }
    pred[sub * 16 + row] = lsum;
    __syncthreads();
    if (tid < 16) {
        float s = 0.0f;
        #pragma unroll
        for (int q = 0; q < 8; ++q) s += pred[q * 16 + tid];
        rowred[tid] = 1.0f / s;
    }
    __syncthreads();

    // --- write normalized tile (rows r0..r0+15, cols 0..999) ---
    const int   r   = tid >> 3;    // 0..15
    const int   cs  = tid & 7;
    const float inv = rowred[r];
    float* orow = out + (size_t)(r0 + r) * C_CLS;
    for (int c = cs; c < C_CLS; c += 8)
        orow[c] = Sc[r * SC_STR + c] * inv;
}

// ---------------------------------------------------------------------------
extern "C" void kernel_launch(void* const* d_in, const int* in_sizes, int n_in,
                              void* d_out, int out_size, void* d_ws, size_t ws_size,
                              hipStream_t stream) {
    const float* logits    = (const float*)d_in[0];
    const float* features  = (const float*)d_in[1];
    const float* mean_vecs = (const float*)d_in[2];
    const float* wb_shape  = (const float*)d_in[3];
    const float* wb_loc    = (const float*)d_in[4];
    const float* wb_scale  = (const float*)d_in[5];
    float* out = (float*)d_out;

    // workspace layout: mT [256][1008] f32 (~1.01 MB), then params [1000] f32x4
    float*  mT     = (float*)d_ws;
    float4* params = (float4*)(mT + (size_t)F_DIM * C_PAD);

    openmax_prep<<<C_PAD, 256, 0, stream>>>(mean_vecs, wb_shape, wb_loc,
                                            wb_scale, mT, params);
    openmax_main<<<B_ROWS / 16, 128, 0, stream>>>(logits, features, mT,
                                                  params, out);
}